// GCN_47150150976048
// MI455X (gfx1250) — compile-verified
//
#include <hip/hip_runtime.h>
#include <hip/hip_bf16.h>

#define F_IN 512
#define HID  16
#define COUT 40

typedef float v2f __attribute__((ext_vector_type(2)));
typedef float v8f __attribute__((ext_vector_type(8)));

// ---------------- utility kernels ----------------
__global__ void fill_f32(float* __restrict__ p, float v, int n) {
    int i = blockIdx.x * blockDim.x + threadIdx.x;
    if (i < n) p[i] = v;
}

// deg[col[e]] += ew[e]   (deg pre-filled with 1.0 for the self loop)
__global__ void deg_kernel(const long long* __restrict__ col,
                           const float* __restrict__ ew,
                           float* __restrict__ deg, int E) {
    int e = blockIdx.x * blockDim.x + threadIdx.x;
    if (e < E) atomicAdd(&deg[col[e]], ew[e]);
}

// deg -> deg^{-1/2} in place
__global__ void dinv_kernel(float* __restrict__ d, int n) {
    int i = blockIdx.x * blockDim.x + threadIdx.x;
    if (i < n) {
        float v = d[i];
        d[i] = (v > 0.0f) ? rsqrtf(v) : 0.0f;
    }
}

// Precompute per-edge int32 endpoints + folded normalization coefficient.
// Cuts per-scatter edge traffic from ~28B (int64 pair + ew + 2 dinv gathers) to 12B.
__global__ void prep_edges(const long long* __restrict__ row,
                           const long long* __restrict__ col,
                           const float* __restrict__ ew,
                           const float* __restrict__ dinv,
                           int2* __restrict__ rc,
                           float* __restrict__ coeff, int E) {
    int e = blockIdx.x * blockDim.x + threadIdx.x;
    if (e >= E) return;
    int r = (int)row[e];
    int c = (int)col[e];
    rc[e] = make_int2(r, c);
    coeff[e] = dinv[r] * ew[e] * dinv[c];
}

// ---------------- WMMA GEMM: out[n x 16] = act(in[n x K] @ W[K x 16] (+bias)) ----------------
// One wave32 computes a 16x16 output tile with V_WMMA_F32_16X16X4_F32.
// A layout (16x4 f32): lanes 0-15 carry rows M=0..15 with K={0,1}; lanes 16-31 same rows, K={2,3}.
// B layout (4x16 f32): lanes 0-15 hold rows K=0 (v0) / K=1 (v1); lanes 16-31 hold K=2 / K=3.
// C/D layout (16x16 f32): vgpr r -> M = (lane>>4)*8 + r, N = lane&15.
template <int K, bool BIAS_RELU>
__global__ void linear16_wmma(const float* __restrict__ in,
                              const float* __restrict__ W,   // K x 16 row-major
                              const float* __restrict__ bias,
                              float* __restrict__ out, int nrows) {
    __shared__ float sW[K * 16];
    for (int i = threadIdx.x; i < K * 16; i += blockDim.x) sW[i] = W[i];
    __syncthreads();

    const int lane = threadIdx.x & 31;
    const int waveInBlock = threadIdx.x >> 5;
    const int tile = blockIdx.x * (blockDim.x >> 5) + waveInBlock; // 16 rows / tile
    const int row0 = tile * 16;
    if (row0 >= nrows) return;               // wave-uniform: EXEC stays all-ones for WMMA

    const int half = lane >> 4;              // 0 or 1
    const int mr   = lane & 15;              // row (for A) / column (for B, D)
    const float* __restrict__ arow = in + (size_t)(row0 + mr) * K + half * 2;

    v8f c = {0.f, 0.f, 0.f, 0.f, 0.f, 0.f, 0.f, 0.f};

    #pragma unroll 4
    for (int kc = 0; kc < K; kc += 4) {
        if (K > 64) {
            // stream the next chunk of this row into near caches (WGP-scope prefetch)
            __builtin_prefetch(arow + kc + 128, 0, 3);
        }
        v2f a = *(const v2f*)(arow + kc);          // A[M=mr][kc+2*half .. +1]
        const int kb = kc + half * 2;
        v2f b;
        b.x = sW[(kb + 0) * 16 + mr];              // B[K=kb  ][N=mr]
        b.y = sW[(kb + 1) * 16 + mr];              // B[K=kb+1][N=mr]
        c = __builtin_amdgcn_wmma_f32_16x16x4_f32(
                false, a, false, b, (short)0, c, false, false);
    }

    const int mbase = half * 8;
    #pragma unroll
    for (int r = 0; r < 8; ++r) {
        float v = c[r];
        if (BIAS_RELU) {
            v += bias[mr];
            v = v > 0.0f ? v : 0.0f;
        }
        out[(size_t)(row0 + mbase + r) * 16 + mr] = v;
    }
}

// ---------------- edge scatter (packed path): acc[col] += coeff * hw[row] ----------------
__global__ void scatter_packed(const int2* __restrict__ rc,
                               const float* __restrict__ coeff,
                               const float* __restrict__ hw,
                               float* __restrict__ acc, int E) {
    int e = blockIdx.x * blockDim.x + threadIdx.x;
    if (e >= E) return;
    int2 p = rc[e];
    float cf = coeff[e];
    const float4* __restrict__ src = (const float4*)(hw + (size_t)p.x * 16);
    float* __restrict__ dst = acc + (size_t)p.y * 16;
    #pragma unroll
    for (int j = 0; j < 4; ++j) {
        float4 v = src[j];
        atomicAdd(dst + 4 * j + 0, cf * v.x);
        atomicAdd(dst + 4 * j + 1, cf * v.y);
        atomicAdd(dst + 4 * j + 2, cf * v.z);
        atomicAdd(dst + 4 * j + 3, cf * v.w);
    }
}

// ---------------- edge scatter (fallback, reads int64 edges directly) ----------------
__global__ void scatter_i64(const long long* __restrict__ row,
                            const long long* __restrict__ col,
                            const float* __restrict__ ew,
                            const float* __restrict__ dinv,
                            const float* __restrict__ hw,
                            float* __restrict__ acc, int E) {
    int e = blockIdx.x * blockDim.x + threadIdx.x;
    if (e >= E) return;
    long long r = row[e];
    long long c = col[e];
    float cf = dinv[r] * ew[e] * dinv[c];
    const float4* __restrict__ src = (const float4*)(hw + (size_t)r * 16);
    float* __restrict__ dst = acc + (size_t)c * 16;
    #pragma unroll
    for (int j = 0; j < 4; ++j) {
        float4 v = src[j];
        atomicAdd(dst + 4 * j + 0, cf * v.x);
        atomicAdd(dst + 4 * j + 1, cf * v.y);
        atomicAdd(dst + 4 * j + 2, cf * v.z);
        atomicAdd(dst + 4 * j + 3, cf * v.w);
    }
}

// h = relu(acc + dinv^2 * hw + bias)   (self-loop term folded in here, no atomics needed)
__global__ void finalize_kernel(const float* __restrict__ acc,
                                const float* __restrict__ hw,
                                const float* __restrict__ dinv,
                                const float* __restrict__ bias,
                                float* __restrict__ hout, int n16) {
    int i = blockIdx.x * blockDim.x + threadIdx.x;
    if (i >= n16) return;
    int node = i >> 4;
    int f = i & 15;
    float s = dinv[node];
    float v = acc[i] + s * s * hw[i] + bias[f];
    hout[i] = v > 0.0f ? v : 0.0f;
}

// ---------------- output layer: logits = h @ W_out + b; log_softmax ----------------
__global__ void out_kernel(const float* __restrict__ h,
                           const float* __restrict__ Wo,  // 16 x 40 row-major
                           const float* __restrict__ bo,
                           float* __restrict__ out, int n) {
    __shared__ float sW[HID * COUT];
    __shared__ float sb[COUT];
    for (int i = threadIdx.x; i < HID * COUT; i += blockDim.x) sW[i] = Wo[i];
    for (int i = threadIdx.x; i < COUT; i += blockDim.x) sb[i] = bo[i];
    __syncthreads();

    int node = blockIdx.x * blockDim.x + threadIdx.x;
    if (node >= n) return;

    float hv[HID];
    const float4* h4 = (const float4*)(h + (size_t)node * HID);
    #pragma unroll
    for (int j = 0; j < 4; ++j) {
        float4 v = h4[j];
        hv[4 * j + 0] = v.x; hv[4 * j + 1] = v.y;
        hv[4 * j + 2] = v.z; hv[4 * j + 3] = v.w;
    }

    float logits[COUT];
    float m = -3.402823466e+38f;
    #pragma unroll
    for (int c = 0; c < COUT; ++c) {
        float a = sb[c];
        #pragma unroll
        for (int f = 0; f < HID; ++f) a = fmaf(hv[f], sW[f * COUT + c], a);
        logits[c] = a;
        m = fmaxf(m, a);
    }
    float sum = 0.0f;
    #pragma unroll
    for (int c = 0; c < COUT; ++c) sum += expf(logits[c] - m);
    float lse = m + logf(sum);
    float* o = out + (size_t)node * COUT;
    #pragma unroll
    for (int c = 0; c < COUT; ++c) o[c] = logits[c] - lse;
}

// ---------------- launch ----------------
extern "C" void kernel_launch(void* const* d_in, const int* in_sizes, int n_in,
                              void* d_out, int out_size, void* d_ws, size_t ws_size,
                              hipStream_t stream) {
    const float*     x       = (const float*)d_in[0];
    const long long* eidx    = (const long long*)d_in[1];  // int64 per reference
    const float*     ew      = (const float*)d_in[2];
    const float*     W_first = (const float*)d_in[3];
    const float*     b_first = (const float*)d_in[4];
    const float*     W1      = (const float*)d_in[5];
    const float*     b1      = (const float*)d_in[6];
    const float*     W2      = (const float*)d_in[7];
    const float*     b2      = (const float*)d_in[8];
    const float*     W_out   = (const float*)d_in[9];
    const float*     b_out   = (const float*)d_in[10];

    const int N = in_sizes[0] / F_IN;
    const int E = in_sizes[1] / 2;
    const long long* row = eidx;
    const long long* col = eidx + E;

    // Workspace layout (8B-aligned region first):
    //   rc    : E int2      (packed edge endpoints)   -- optional
    //   coeff : E float     (folded norm coefficient) -- optional
    //   dinv  : N float
    //   h     : 16N float
    //   hw    : 16N float
    //   acc   : 16N float
    const size_t baseFloats = (size_t)49 * N;                 // dinv + h + hw + acc
    const size_t packedBytes = (size_t)E * 12 + baseFloats * 4;
    const bool usePacked = ws_size >= packedBytes;

    char* wsb = (char*)d_ws;
    int2*  rc    = nullptr;
    float* coeff = nullptr;
    float* fbase;
    if (usePacked) {
        rc    = (int2*)wsb;
        coeff = (float*)(wsb + (size_t)E * 8);
        fbase = (float*)(wsb + (size_t)E * 12);
    } else {
        fbase = (float*)wsb;
    }
    float* dinv = fbase;                 // N
    float* h    = dinv + N;              // 16N
    float* hw   = h + (size_t)16 * N;    // 16N
    float* acc  = hw + (size_t)16 * N;   // 16N

    const int TB = 256;
    // ----- degree / normalization -----
    fill_f32<<<(N + TB - 1) / TB, TB, 0, stream>>>(dinv, 1.0f, N);          // self-loop weight
    deg_kernel<<<(E + TB - 1) / TB, TB, 0, stream>>>(col, ew, dinv, E);
    dinv_kernel<<<(N + TB - 1) / TB, TB, 0, stream>>>(dinv, N);
    if (usePacked) {
        prep_edges<<<(E + TB - 1) / TB, TB, 0, stream>>>(row, col, ew, dinv, rc, coeff, E);
    }

    // ----- first linear: h = relu(x @ W_first + b_first) -----
    const int tiles = (N + 15) / 16;
    const int wavesPerBlock = TB / 32;
    dim3 gemmGrid((tiles + wavesPerBlock - 1) / wavesPerBlock);
    linear16_wmma<F_IN, true><<<gemmGrid, TB, 0, stream>>>(x, W_first, b_first, h, N);

    // ----- two GCNConv layers -----
    const float* Ws[2] = {W1, W2};
    const float* bs[2] = {b1, b2};
    for (int layer = 0; layer < 2; ++layer) {
        linear16_wmma<HID, false><<<gemmGrid, TB, 0, stream>>>(h, Ws[layer], nullptr, hw, N);
        fill_f32<<<(16 * N + TB - 1) / TB, TB, 0, stream>>>(acc, 0.0f, 16 * N);
        if (usePacked) {
            scatter_packed<<<(E + TB - 1) / TB, TB, 0, stream>>>(rc, coeff, hw, acc, E);
        } else {
            scatter_i64<<<(E + TB - 1) / TB, TB, 0, stream>>>(row, col, ew, dinv, hw, acc, E);
        }
        finalize_kernel<<<(16 * N + TB - 1) / TB, TB, 0, stream>>>(acc, hw, dinv, bs[layer], h, 16 * N);
    }

    // ----- output layer + log_softmax -----
    out_kernel<<<(N + TB - 1) / TB, TB, 0, stream>>>(h, W_out, b_out, (float*)d_out, N);
}